// FastHadamardTransform_80101140070866
// MI455X (gfx1250) — compile-verified
//
#include <hip/hip_runtime.h>

// FWHT-4096 via H16 x H16 x H16 Kronecker decomposition on CDNA5 WMMA.
// One workgroup (256 threads = 8 wave32) per row; row lives in padded LDS.

typedef __attribute__((ext_vector_type(2))) float v2f;
typedef __attribute__((ext_vector_type(8))) float v8f;

// Pad 1 float per 16 to kill bank conflicts on stride-16 accesses.
#define LP(i) ((i) + ((i) >> 4))

__device__ __forceinline__ float hsign(int r, int c) {
    return (__builtin_popcount(r & c) & 1) ? -1.0f : 1.0f;
}

__device__ __forceinline__ v8f h16_gemm(const v2f A[4], const v2f B[4]) {
    v8f C = {};
#pragma unroll
    for (int kk = 0; kk < 4; ++kk) {
        // 8 args: (neg_a, A, neg_b, B, c_mod, C, reuse_a, reuse_b)
        C = __builtin_amdgcn_wmma_f32_16x16x4_f32(
            false, A[kk], false, B[kk], (short)0, C, false, false);
    }
    return C;
}

__global__ __launch_bounds__(256)
void fwht4096_wmma_kernel(const float* __restrict__ x,
                          const float* __restrict__ scale,
                          float* __restrict__ out) {
    __shared__ float lds[4096 + 256];

    const int tid  = threadIdx.x;
    const int lane = tid & 31;
    const int wave = tid >> 5;
    const int m    = lane & 15;   // N column (B/C) / M row (A)
    const int hi   = lane >> 4;   // half-wave select
    const size_t rowbase = (size_t)blockIdx.x * 4096;

    // ---- coalesced global -> padded LDS (16 floats per thread) ----
    const float4* xin = (const float4*)(x + rowbase);
#pragma unroll
    for (int c = 0; c < 4; ++c) {
        int idx = c * 1024 + tid * 4;           // multiple of 4, within one 16-group
        float4 v = xin[idx >> 2];
        int a = LP(idx);
        lds[a + 0] = v.x; lds[a + 1] = v.y; lds[a + 2] = v.z; lds[a + 3] = v.w;
    }

    // ---- A operand = H16, split into four 16x4 K-chunks (constant all modes) ----
    v2f A[4];
#pragma unroll
    for (int kk = 0; kk < 4; ++kk) {
        int k0 = 4 * kk + 2 * hi;
        A[kk].x = hsign(m, k0);
        A[kk].y = hsign(m, k0 + 1);
    }

    const float s = scale[0];
    __syncthreads();

    // ================= mode-k: y1[p*16 + k'] = sum_k H[k',k] x[p*16 + k] =========
    // tile t: B[k][n] = x[(t*16+n)*16 + k], in-place per-tile update
#pragma unroll
    for (int tt = 0; tt < 2; ++tt) {
        int t = wave * 2 + tt;
        int base = (t * 16 + m) * 16;
        v2f B[4];
#pragma unroll
        for (int kk = 0; kk < 4; ++kk) {
            int k0 = 4 * kk + 2 * hi;
            B[kk].x = lds[LP(base + k0)];
            B[kk].y = lds[LP(base + k0 + 1)];
        }
        v8f C = h16_gemm(A, B);
#pragma unroll
        for (int p = 0; p < 8; ++p)
            lds[LP(base + p + 8 * hi)] = C[p];
    }
    __syncthreads();

    // ================= mode-j: tile i=t: B[j][k] = y1[t*256 + j*16 + k] ==========
#pragma unroll
    for (int tt = 0; tt < 2; ++tt) {
        int t = wave * 2 + tt;
        v2f B[4];
#pragma unroll
        for (int kk = 0; kk < 4; ++kk) {
            int j0 = 4 * kk + 2 * hi;
            B[kk].x = lds[LP(t * 256 + j0 * 16 + m)];
            B[kk].y = lds[LP(t * 256 + (j0 + 1) * 16 + m)];
        }
        v8f C = h16_gemm(A, B);
#pragma unroll
        for (int p = 0; p < 8; ++p)
            lds[LP(t * 256 + (p + 8 * hi) * 16 + m)] = C[p];
    }
    __syncthreads();

    // ============ mode-i: tile t of jk-columns: B[i][n] = y2[i*256 + t*16 + n] ===
    // C rows are i'; write straight to global with scale (64B per half-wave).
#pragma unroll
    for (int tt = 0; tt < 2; ++tt) {
        int t = wave * 2 + tt;
        v2f B[4];
#pragma unroll
        for (int kk = 0; kk < 4; ++kk) {
            int i0 = 4 * kk + 2 * hi;
            B[kk].x = lds[LP(i0 * 256 + t * 16 + m)];
            B[kk].y = lds[LP((i0 + 1) * 256 + t * 16 + m)];
        }
        v8f C = h16_gemm(A, B);
#pragma unroll
        for (int p = 0; p < 8; ++p)
            out[rowbase + (size_t)(p + 8 * hi) * 256 + t * 16 + m] = C[p] * s;
    }
}

extern "C" void kernel_launch(void* const* d_in, const int* in_sizes, int n_in,
                              void* d_out, int out_size, void* d_ws, size_t ws_size,
                              hipStream_t stream) {
    (void)in_sizes; (void)n_in; (void)d_ws; (void)ws_size;
    const float* x     = (const float*)d_in[0];
    const float* scale = (const float*)d_in[1];
    float*       out   = (float*)d_out;
    const int rows = out_size / 4096;   // 16384 rows of length 4096
    fwht4096_wmma_kernel<<<dim3(rows), dim3(256), 0, stream>>>(x, scale, out);
}